// HoughVotingLayer_65790309040389
// MI455X (gfx1250) — compile-verified
//
#include <hip/hip_runtime.h>
#include <hip/hip_bf16.h>
#include <math.h>

// ---------------- problem constants ----------------
#define HH 480
#define WW 640
#define NPX 128            // W/SKIP
#define NPY 96             // H/SKIP
#define NP (NPX*NPY)       // 12288 sample points
#define NBX 80             // W/GRID
#define NBY 60             // H/GRID
#define NB (NBX*NBY)       // 4800 bins
#define NTI (NP/16)        // 768 i-tiles
#define NTJ (NB/16)        // 300 j-tiles
#define ISLICES 16
#define TPS (NTI/ISLICES)  // 48 i-tiles per wave

typedef float v2f __attribute__((ext_vector_type(2)));
typedef float v8f __attribute__((ext_vector_type(8)));

// D = A(16x4, f32) * B(4x16, f32) + C(16x16, f32); full-precision WMMA.
__device__ __forceinline__ v8f wmma4(v2f a, v2f b, v8f c) {
  return __builtin_amdgcn_wmma_f32_16x16x4_f32(false, a, false, b, (short)0, c, false, false);
}

// ---------------- phase 0: zero accumulators ----------------
__global__ void zero_kernel(float* hough, float* dsum, float* counts) {
  int idx = blockIdx.x * blockDim.x + threadIdx.x;
  if (idx < 3 * NB) { hough[idx] = 0.f; dsum[idx] = 0.f; }
  if (idx < 3) counts[idx] = 0.f;
}

// ---------------- phase 1: per-point operands ----------------
// pA[i] = (ux, uy, cterm, dist)   cterm = -(vx*ux + vy*uy)
// pB[i] = (-2vx, -2vy, vx^2+vy^2+1e-6, lbl)
__global__ void prep_kernel(const int* __restrict__ label,
                            const float* __restrict__ vert,
                            float4* __restrict__ pA, float4* __restrict__ pB,
                            float* __restrict__ dist, float* __restrict__ counts) {
  int i = blockIdx.x * blockDim.x + threadIdx.x;
  if (i >= NP) return;
  int yi = i / NPX, xi = i % NPX;
  int vy = yi * 5, vx = xi * 5;
  int pix = vy * WW + vx;
  int lbl = label[pix];
  const float* base = vert + (size_t)lbl * 3 * HH * WW + pix;
  float ux = base[0];
  float uy = base[(size_t)HH * WW];
  float dz = base[2 * (size_t)HH * WW];
  float fvx = (float)vx, fvy = (float)vy;
  float4 a; a.x = ux; a.y = uy; a.z = -(fvx * ux + fvy * uy); a.w = dz;
  float4 b; b.x = -2.f * fvx; b.y = -2.f * fvy;
  b.z = fvx * fvx + fvy * fvy + 1e-6f; b.w = (float)lbl;
  pA[i] = a; pB[i] = b; dist[i] = dz;
  atomicAdd(&counts[lbl], 1.0f);
}

// ---------------- phase 1b: pack 8 labels (4 bits each) per half-tile ----------------
__global__ void pack_kernel(const int* __restrict__ label, unsigned* __restrict__ lblPack) {
  int w = blockIdx.x * blockDim.x + threadIdx.x;
  if (w >= NTI * 2) return;
  int base = (w >> 1) * 16 + (w & 1) * 8;
  unsigned p = 0;
  for (int r = 0; r < 8; ++r) {
    int i = base + r;
    int yi = i / NPX, xi = i % NPX;
    unsigned lbl = (unsigned)label[(yi * 5) * WW + xi * 5];
    p |= (lbl & 0xFu) << (4 * r);
  }
  lblPack[w] = p;
}

// ---------------- phase 2: WMMA voting ----------------
// One wave (32 lanes) per (j-tile, i-slice). A layout: lanes 0-15 -> K=0,1;
// lanes 16-31 -> K=2,3. D layout: reg r, lane L -> M = r + 8*(L>=16), N = L%16.
__global__ __launch_bounds__(32) void vote_kernel(
    const float4* __restrict__ pA, const float4* __restrict__ pB,
    const float* __restrict__ dist, const unsigned* __restrict__ lblPack,
    float* __restrict__ hough, float* __restrict__ dsum) {
  int jt = blockIdx.x, slice = blockIdx.y;
  int lane = threadIdx.x;
  int half = lane >> 4;
  int lj = lane & 15;
  int j = jt * 16 + lj;
  float fbx = 4.f + 8.f * (float)(j % NBX);
  float fby = 4.f + 8.f * (float)(j / NBX);
  float rr = fbx * fbx + fby * fby;

  // B columns (fixed per wave): num: [bx; by; 1; 0]   n2: [bx^2+by^2; bx; by; 1]
  v2f Bnum, Bn2;
  Bnum[0] = half ? 1.f : fbx;  Bnum[1] = half ? 0.f : fby;
  Bn2[0]  = half ? fby : rr;   Bn2[1]  = half ? 1.f : fbx;

  float h1 = 0.f, h2 = 0.f, d1 = 0.f, d2 = 0.f;
  int t0 = slice * TPS;
  for (int t = t0; t < t0 + TPS; ++t) {
    int ibase = t * 16;
    float4 a = pA[ibase + lj];
    float4 b = pB[ibase + lj];
    // A rows: num: [ux, uy, cterm, 0]   n2: [1, -2vx, -2vy, q]
    v2f Anum, An2;
    Anum[0] = half ? a.z : a.x;  Anum[1] = half ? 0.f : a.y;
    An2[0]  = half ? b.y : 1.f;  An2[1]  = half ? b.z : b.x;

    v8f cz = {0.f, 0.f, 0.f, 0.f, 0.f, 0.f, 0.f, 0.f};
    v8f num = wmma4(Anum, Bnum, cz);
    v8f n2  = wmma4(An2,  Bn2,  cz);

    // dists + labels for this lane's 8 rows (rows half*8 .. half*8+7)
    const float4* dp = (const float4*)(dist + ibase + half * 8);
    float4 da = dp[0], db = dp[1];
    float dd[8] = {da.x, da.y, da.z, da.w, db.x, db.y, db.z, db.w};
    unsigned lp = lblPack[t * 2 + half];

#pragma unroll
    for (int r = 0; r < 8; ++r) {
      float nv = num[r], qv = n2[r];
      // cos > 0.5  <=>  num > 0  &&  num^2 > 0.25 * n2   (rsqrt-free)
      float wv = (nv > 0.f && nv * nv > 0.25f * qv) ? 1.f : 0.f;
      int lb = (int)((lp >> (4 * r)) & 0xFu);
      float w1 = (lb == 1) ? wv : 0.f;
      float w2 = (lb == 2) ? wv : 0.f;
      h1 += w1; h2 += w2;
      d1 += w1 * dd[r]; d2 += w2 * dd[r];
    }
  }
  // combine the two M-halves holding the same column j
  h1 += __shfl_xor(h1, 16, 32);
  h2 += __shfl_xor(h2, 16, 32);
  d1 += __shfl_xor(d1, 16, 32);
  d2 += __shfl_xor(d2, 16, 32);
  if (lane < 16) {
    atomicAdd(&hough[1 * NB + j], h1);
    atomicAdd(&hough[2 * NB + j], h2);
    atomicAdd(&dsum[1 * NB + j], d1);
    atomicAdd(&dsum[2 * NB + j], d2);
  }
}

// ---------------- phase 3: argmax + ROI epilogue ----------------
__global__ __launch_bounds__(256) void finalize_kernel(
    const float* __restrict__ hough, const float* __restrict__ dsum,
    const float* __restrict__ counts, const float* __restrict__ meta,
    const float* __restrict__ extents, float* __restrict__ rois) {
  __shared__ float sv[256];
  __shared__ int si[256];
  int tid = threadIdx.x;
  for (int c = 0; c < 3; ++c) {
    float best = -1.f; int bidx = 0;
    for (int j = tid; j < NB; j += 256) {
      float v = hough[c * NB + j];
      if (v > best) { best = v; bidx = j; }   // strict > keeps lowest j per thread
    }
    sv[tid] = best; si[tid] = bidx;
    __syncthreads();
    for (int s = 128; s > 0; s >>= 1) {
      if (tid < s) {
        float ov = sv[tid + s]; int oi = si[tid + s];
        if (ov > sv[tid] || (ov == sv[tid] && oi < si[tid])) { sv[tid] = ov; si[tid] = oi; }
      }
      __syncthreads();
    }
    if (tid == 0) {
      int peak = si[0];
      float votes = sv[0];
      float depth = dsum[c * NB + peak] / fmaxf(votes, 1.f);
      float cx = 4.f + 8.f * (float)(peak % NBX);
      float cy = 4.f + 8.f * (float)(peak / NBX);
      float cnt = counts[c];
      float score = votes / fmaxf(cnt, 1.f);
      bool valid = (cnt > 5.f) && (score > 0.3f);
      float fx = meta[0], fy = meta[4];
      float e0 = extents[c * 3 + 0], e1 = extents[c * 3 + 1], e2 = extents[c * 3 + 2];
      float diag = sqrtf(e0 * e0 + e1 * e1 + e2 * e2);
      float safez = fmaxf(fabsf(depth), 0.001f);
      float bw = fabsf(diag * fx) / safez;
      float bh = fabsf(diag * fy) / safez;
      rois[c * 6 + 0] = (float)c;
      rois[c * 6 + 1] = cx - bw * 0.5f;
      rois[c * 6 + 2] = cy - bh * 0.5f;
      rois[c * 6 + 3] = cx + bw * 0.5f;
      rois[c * 6 + 4] = cy + bh * 0.5f;
      rois[c * 6 + 5] = valid ? score : 0.f;
    }
    __syncthreads();
  }
}

extern "C" void kernel_launch(void* const* d_in, const int* in_sizes, int n_in,
                              void* d_out, int out_size, void* d_ws, size_t ws_size,
                              hipStream_t stream) {
  (void)in_sizes; (void)n_in; (void)out_size; (void)ws_size;
  const int*   label   = (const int*)d_in[0];
  const float* vert    = (const float*)d_in[1];
  const float* meta    = (const float*)d_in[2];
  const float* extents = (const float*)d_in[3];

  float* out   = (float*)d_out;
  float* rois  = out;        // 3 x 6
  float* hough = out + 18;   // 3 x 4800 (accumulated in place)

  // workspace layout (floats): pA[NP*4] pB[NP*4] dist[NP] lblPack[NTI*2] dsum[3*NB] counts[3]
  float* ws = (float*)d_ws;
  float4*   pA      = (float4*)ws;
  float4*   pB      = (float4*)(ws + NP * 4);
  float*    dist    = ws + NP * 8;
  unsigned* lblPack = (unsigned*)(ws + NP * 9);
  float*    dsum    = ws + NP * 9 + NTI * 2;
  float*    counts  = dsum + 3 * NB;

  zero_kernel<<<(3 * NB + 255) / 256, 256, 0, stream>>>(hough, dsum, counts);
  prep_kernel<<<(NP + 255) / 256, 256, 0, stream>>>(label, vert, pA, pB, dist, counts);
  pack_kernel<<<(NTI * 2 + 255) / 256, 256, 0, stream>>>(label, lblPack);
  vote_kernel<<<dim3(NTJ, ISLICES), 32, 0, stream>>>(pA, pB, dist, lblPack, hough, dsum);
  finalize_kernel<<<1, 256, 0, stream>>>(hough, dsum, counts, meta, extents, rois);
}